// QuantumLayer_13778255085939
// MI455X (gfx1250) — compile-verified
//
#include <hip/hip_runtime.h>
#include <math.h>

// out[b] = F * cos(x[b,0])
// F = (cos^2(w0/2)-sin^2(w0/2)) * (cos^2(w1/2)+sin^2(w1/2))
// (CNOT and qubit-1 terms cancel in <Z0> to within ~1 ulp; see prior analysis.)

#ifndef QL_UNROLL
#define QL_UNROLL 4
#endif

#if defined(__HIP_DEVICE_COMPILE__) &&                                   \
    __has_builtin(__builtin_amdgcn_global_load_async_to_lds_b128) &&     \
    __has_builtin(__builtin_amdgcn_s_wait_asynccnt)
#define QL_ASYNC 1
#else
#define QL_ASYNC 0
#endif

#if QL_ASYNC
typedef int qv4i __attribute__((vector_size(16)));
#define QL_GLOBAL __attribute__((address_space(1)))
#define QL_LDS    __attribute__((address_space(3)))
#endif

__global__ __launch_bounds__(32)
void qlayer_factor_kernel(const float* __restrict__ w, float* __restrict__ f) {
    if (threadIdx.x == 0) {
        float hw0 = 0.5f * w[0];
        float hw1 = 0.5f * w[1];
        float cw0 = cosf(hw0), sw0 = sinf(hw0);
        float cw1 = cosf(hw1), sw1 = sinf(hw1);
        f[0] = (cw0 * cw0 - sw0 * sw0) * (cw1 * cw1 + sw1 * sw1);
    }
}

// x4[k] = {x[2k,0], x[2k,1], x[2k+1,0], x[2k+1,1]} -> out2[k] = {z0[2k], z0[2k+1]}
__global__ __launch_bounds__(256)
void qlayer_z0_kernel(const float4* __restrict__ x4,
                      const float*  __restrict__ fp,
                      float2*       __restrict__ out2,
                      int npairs) {
    const int T = gridDim.x * blockDim.x;          // total threads
    const int t = blockIdx.x * blockDim.x + threadIdx.x;
    const float f = fp[0];                         // uniform -> scalar load

#if QL_ASYNC
    // gfx1250 async staging: each lane owns one 16B LDS slot per buffer.
    // Double-buffered GLOBAL_LOAD_ASYNC_TO_LDS_B128, tracked by ASYNCcnt.
    __shared__ float4 tile[2][256];                // 8 KB of the 320 KB/WGP

    const int last = (npairs > 0) ? (npairs - 1) : 0;

    auto issue = [&](int i, int buf) {
        int idx = t + i * T;
        if (idx > last) idx = last;                // clamp: every lane always
                                                   // issues -> exact ASYNCcnt
        __builtin_amdgcn_global_load_async_to_lds_b128(
            (QL_GLOBAL qv4i*)(x4 + idx),
            (QL_LDS qv4i*)&tile[buf][threadIdx.x],
            /*offset=*/0, /*cpol=*/0);
    };

    auto stage = [&](int i, int buf) {
        asm volatile("" ::: "memory");             // don't hoist the ds_load
        const float4 v = tile[buf][threadIdx.x];   // ds_load_b128 (own slot;
                                                   // wave-level wait suffices)
        const int idx = t + i * T;
        if (idx < npairs) {
            float2 r;
            r.x = f * cosf(v.x);
            r.y = f * cosf(v.z);
            out2[idx] = r;                         // global_store_b64
        }
    };

    // 4-stage software pipeline, depth 2; wait immediates must be literals.
    issue(0, 0);
    issue(1, 1); __builtin_amdgcn_s_wait_asynccnt(1); stage(0, 0);
    issue(2, 0); __builtin_amdgcn_s_wait_asynccnt(1); stage(1, 1);
    issue(3, 1); __builtin_amdgcn_s_wait_asynccnt(1); stage(2, 0);
                 __builtin_amdgcn_s_wait_asynccnt(0); stage(3, 1);
#else
    // Fallback: direct b128 loads + gfx1250 global_prefetch_b8 look-ahead.
#pragma unroll
    for (int i = 0; i < QL_UNROLL; ++i) {
        const int idx = t + i * T;
        if (i + 1 < QL_UNROLL) {
            // Speculative prefetch: OOB translation failures silently dropped.
            __builtin_prefetch((const void*)(x4 + idx + T), 0, 0);
        }
        if (idx < npairs) {
            const float4 v = x4[idx];              // global_load_b128
            float2 r;
            r.x = f * cosf(v.x);
            r.y = f * cosf(v.z);
            out2[idx] = r;                         // global_store_b64
        }
    }
#endif
}

extern "C" void kernel_launch(void* const* d_in, const int* in_sizes, int n_in,
                              void* d_out, int out_size, void* d_ws, size_t ws_size,
                              hipStream_t stream) {
    const float* x = (const float*)d_in[0];   // [B, 2] float32
    const float* w = (const float*)d_in[1];   // [2]    float32
    float* out = (float*)d_out;               // [B, 1] float32
    float* ws  = (float*)d_ws;                // ws[0] = scalar factor F

    // 1) scalar factor from weights (one wave; stream order gives the dep).
    qlayer_factor_kernel<<<1, 32, 0, stream>>>(w, ws);

    // 2) streaming map: each float4 chunk = 2 batch elements.
    const int npairs = in_sizes[0] / 4;       // 4,194,304 for B = 8,388,608
    const int BLK = 256;
    const int threads_total = (npairs + QL_UNROLL - 1) / QL_UNROLL;
    const int blocks = (threads_total + BLK - 1) / BLK;   // 4096
    qlayer_z0_kernel<<<blocks, BLK, 0, stream>>>(
        (const float4*)x, ws, (float2*)out, npairs);
}